// SRA_53034256171445
// MI455X (gfx1250) — compile-verified
//
#include <hip/hip_runtime.h>

typedef __attribute__((ext_vector_type(2))) float v2f;
typedef __attribute__((ext_vector_type(4))) float v4f;
typedef __attribute__((ext_vector_type(8))) float v8f;
typedef __attribute__((ext_vector_type(4))) unsigned int v4u;
typedef __attribute__((ext_vector_type(4))) int v4i;
typedef __attribute__((ext_vector_type(8))) int v8i;

namespace {
constexpr int kB    = 16;
constexpr int kC    = 64;
constexpr int kH    = 64;
constexpr int kW    = 64;
constexpr int kN    = kH * kW;              // 4096 query positions
constexpr int kNkv  = (kH / 2) * (kW / 2);  // 1024 key positions
constexpr int kHeads = 2;
constexpr int kHd    = 32;
constexpr int kVST   = 33;                  // padded ldsVt row stride (floats)
constexpr int kPST   = 66;                  // padded P-tile row stride (floats)
constexpr float kScale = 0.17677669529663687f;  // 32^-0.5
}

// ---- VALU 16-lane butterfly reductions via DPP16 ROW_XMASK (no DS pipe) ----
template <int CTRL>
__device__ inline float dpp_movf(float x) {
  return __builtin_bit_cast(
      float, __builtin_amdgcn_update_dpp(0, __builtin_bit_cast(int, x), CTRL,
                                         0xF, 0xF, true));
}
__device__ inline float rowmax16(float x) {
  x = fmaxf(x, dpp_movf<0x161>(x));  // ROW_XMASK:1
  x = fmaxf(x, dpp_movf<0x162>(x));  // ROW_XMASK:2
  x = fmaxf(x, dpp_movf<0x164>(x));  // ROW_XMASK:4
  x = fmaxf(x, dpp_movf<0x168>(x));  // ROW_XMASK:8
  return x;
}
__device__ inline float rowsum16(float x) {
  x += dpp_movf<0x161>(x);
  x += dpp_movf<0x162>(x);
  x += dpp_movf<0x164>(x);
  x += dpp_movf<0x168>(x);
  return x;
}

// ---------------------------------------------------------------------------
// 1x1 conv over [B, 64, 4096]: weight reads are wave-uniform -> scalar loads.
// ---------------------------------------------------------------------------
__global__ __launch_bounds__(256) void sra_conv1x1(const float* __restrict__ in,
                                                   const float* __restrict__ w,
                                                   float* __restrict__ out) {
  const int b = blockIdx.y;
  const int n = blockIdx.x * 256 + threadIdx.x;
  const float* xb = in + (size_t)b * kC * kN + n;
  float acc[kC];
#pragma unroll
  for (int c = 0; c < kC; ++c) acc[c] = 0.f;
  for (int ci = 0; ci < kC; ++ci) {
    const float xv = xb[(size_t)ci * kN];
#pragma unroll
    for (int c = 0; c < kC; ++c) acc[c] = fmaf(w[c * kC + ci], xv, acc[c]);
  }
  float* ob = out + (size_t)b * kC * kN + n;
#pragma unroll
  for (int c = 0; c < kC; ++c) ob[(size_t)c * kN] = acc[c];
}

// ---------------------------------------------------------------------------
// 2x2 stride-2 conv: out[c, m] (m over 32x32)
// ---------------------------------------------------------------------------
__global__ __launch_bounds__(256) void sra_conv2x2s2(const float* __restrict__ in,
                                                     const float* __restrict__ w,
                                                     float* __restrict__ out) {
  const int b = blockIdx.y;
  const int m = blockIdx.x * 256 + threadIdx.x;
  const int ky = m >> 5, kx = m & 31;
  const float* xb = in + (size_t)b * kC * kN + (2 * ky) * kW + 2 * kx;
  float acc[kC];
#pragma unroll
  for (int c = 0; c < kC; ++c) acc[c] = 0.f;
  for (int ci = 0; ci < kC; ++ci) {
    const float* xp = xb + (size_t)ci * kN;
    const float x00 = xp[0], x01 = xp[1], x10 = xp[kW], x11 = xp[kW + 1];
#pragma unroll
    for (int c = 0; c < kC; ++c) {
      const float* wc = w + (((size_t)c * kC + ci) << 2);
      float a = acc[c];
      a = fmaf(wc[0], x00, a);
      a = fmaf(wc[1], x01, a);
      a = fmaf(wc[2], x10, a);
      a = fmaf(wc[3], x11, a);
      acc[c] = a;
    }
  }
  float* ob = out + (size_t)b * kC * kNkv + m;
#pragma unroll
  for (int c = 0; c < kC; ++c) ob[(size_t)c * kNkv] = acc[c];
}

// ---------------------------------------------------------------------------
// Fused flash-style attention per (b, head). 8 waves x 16 queries each.
// K staged into LDS by the Tensor Data Mover; V staged transposed (+padded)
// by the ALUs in parallel. Online softmax over 64-key blocks with DPP
// butterflies. All matmuls: V_WMMA_F32_16X16X4_F32 (fp32-exact).
// ---------------------------------------------------------------------------
__global__ __launch_bounds__(256) void sra_attn(const float* __restrict__ Qg,
                                                const float* __restrict__ Kg,
                                                const float* __restrict__ Vg,
                                                float* __restrict__ Ao) {
  extern __shared__ float smem[];
  float* ldsK  = smem;                        // [32][1024]
  float* ldsVt = smem + kHd * kNkv;           // [1024][33] transposed V
  float* ldsP  = ldsVt + kNkv * kVST;         // 8 waves * 16 * kPST

  const int tid  = threadIdx.x;
  const int wave = tid >> 5;
  const int lane = tid & 31;
  const int col  = lane & 15;
  const int hi   = lane >> 4;

  const int qt   = blockIdx.x;
  const int head = blockIdx.y;
  const int b    = blockIdx.z;

  const size_t kvbase = ((size_t)b * kC + head * kHd) * (size_t)kNkv;

  // ---- stage K via TDM: D# = 2-D tensor 1024 x 32 (f32), tile = all of it
  if (wave == 0) {
    const unsigned long long ga = (unsigned long long)(Kg + kvbase);
    const unsigned ldsa = (unsigned)(unsigned long long)(uintptr_t)ldsK;
    v4u g0;
    g0[0] = 1u;                                   // count=1, user D#
    g0[1] = ldsa;                                 // lds_addr (bytes)
    g0[2] = (unsigned)(ga & 0xFFFFFFFFull);       // global_addr[31:0]
    g0[3] = (unsigned)((ga >> 32) & 0x01FFFFFFull) | (2u << 30);  // type=2
    v8i g1;
    g1[0] = (int)(2u << 16);        // data_size = 4B
    g1[1] = (int)(1024u << 16);     // tensor_dim0 = 1024 (low 16 in [63:48])
    g1[2] = (int)(32u << 16);       // tensor_dim1 = 32
    g1[3] = (int)(1024u << 16);     // tile_dim0 = 1024
    g1[4] = 32;                     // tile_dim1 = 32, tile_dim2 = 0
    g1[5] = 1024;                   // tensor_dim0_stride = 1024
    g1[6] = 0;
    g1[7] = 0;
    v4i z4 = {0, 0, 0, 0};
    v8i z8 = {0, 0, 0, 0, 0, 0, 0, 0};
    __builtin_amdgcn_tensor_load_to_lds(g0, g1, z4, z4, z8, 0);
  }

  // ---- meanwhile: stage V transposed into padded LDS (conflict-free writes)
  for (int idx = tid; idx < kHd * kNkv; idx += 256) {
    const int d = idx >> 10;        // 0..31
    const int m = idx & (kNkv - 1); // consecutive per lane -> coalesced read
    ldsVt[m * kVST + d] = Vg[kvbase + idx];
  }

  if (wave == 0) __builtin_amdgcn_s_wait_tensorcnt(0);
  __syncthreads();

  const int q0 = qt * 128 + wave * 16;
  const size_t qbase = ((size_t)b * kC + head * kHd) * (size_t)kN;

  // ---- preload Q A-fragments (A 16x4 f32: lane -> M=l&15, K=2*(l>>4)+j)
  v2f aq[8];
#pragma unroll
  for (int dc = 0; dc < 8; ++dc) {
    const int d = dc * 4 + 2 * hi;
    aq[dc].x = Qg[qbase + (size_t)d * kN + q0 + col];
    aq[dc].y = Qg[qbase + (size_t)(d + 1) * kN + q0 + col];
  }

  v8f o0 = {}, o1 = {};
  float mrow[8], lrow[8];
#pragma unroll
  for (int r = 0; r < 8; ++r) { mrow[r] = -1e30f; lrow[r] = 0.f; }

  float* pb = ldsP + wave * 16 * kPST;  // wave-private 16 x 64 P tile (padded)

  for (int m0 = 0; m0 < kNkv; m0 += 64) {
    // ---- four S tiles = (Q tile)^T * K[., m0..m0+63]
    v8f s[4];
#pragma unroll
    for (int t = 0; t < 4; ++t) {
      v8f acc = {};
#pragma unroll
      for (int dc = 0; dc < 8; ++dc) {
        const int d = dc * 4 + 2 * hi;
        v2f bk;
        bk.x = ldsK[d * kNkv + m0 + t * 16 + col];
        bk.y = ldsK[(d + 1) * kNkv + m0 + t * 16 + col];
        acc = __builtin_amdgcn_wmma_f32_16x16x4_f32(false, aq[dc], false, bk,
                                                    (short)0, acc, false, false);
      }
      s[t] = acc;
    }

    // ---- online softmax over the 64-key block (one reduction pair per row)
#pragma unroll
    for (int r = 0; r < 8; ++r) {
      const float v0 = s[0][r] * kScale, v1 = s[1][r] * kScale;
      const float v2 = s[2][r] * kScale, v3 = s[3][r] * kScale;
      const float rm = rowmax16(fmaxf(fmaxf(v0, v1), fmaxf(v2, v3)));
      const float mnew  = fmaxf(mrow[r], rm);
      const float alpha = __expf(mrow[r] - mnew);
      const float p0 = __expf(v0 - mnew), p1 = __expf(v1 - mnew);
      const float p2 = __expf(v2 - mnew), p3 = __expf(v3 - mnew);
      const float rs = rowsum16((p0 + p1) + (p2 + p3));
      lrow[r] = lrow[r] * alpha + rs;
      mrow[r] = mnew;
      o0[r] *= alpha;
      o1[r] *= alpha;
      float* pr = pb + (r + 8 * hi) * kPST + col;   // C-layout -> row-major P
      pr[0] = p0; pr[16] = p1; pr[32] = p2; pr[48] = p3;
    }
    // Same-wave DS ops are in-order: reads below see the writes above.

    // ---- O += P * V^T (transposed, padded V: conflict-free B-frag reads)
#pragma unroll
    for (int t = 0; t < 4; ++t) {
#pragma unroll
      for (int mc = 0; mc < 4; ++mc) {
        const int ml = t * 16 + mc * 4 + 2 * hi;
        const int mm = m0 + ml;
        v2f ap, bv0, bv1;
        ap.x  = pb[col * kPST + ml];
        ap.y  = pb[col * kPST + ml + 1];
        bv0.x = ldsVt[mm * kVST + col];
        bv0.y = ldsVt[(mm + 1) * kVST + col];
        bv1.x = ldsVt[mm * kVST + 16 + col];
        bv1.y = ldsVt[(mm + 1) * kVST + 16 + col];
        o0 = __builtin_amdgcn_wmma_f32_16x16x4_f32(false, ap, false, bv0,
                                                   (short)0, o0, false, false);
        o1 = __builtin_amdgcn_wmma_f32_16x16x4_f32(false, ap, false, bv1,
                                                   (short)0, o1, false, false);
      }
    }
  }

  // ---- normalize and store (each lane's 8 C rows are consecutive n)
#pragma unroll
  for (int r = 0; r < 8; ++r) {
    const float inv = 1.f / lrow[r];
    o0[r] *= inv;
    o1[r] *= inv;
  }
  {
    float* p0 = Ao + qbase + (size_t)col * kN + q0 + 8 * hi;
    float* p1 = Ao + qbase + (size_t)(16 + col) * kN + q0 + 8 * hi;
    v4f t;
    t.x = o0[0]; t.y = o0[1]; t.z = o0[2]; t.w = o0[3]; *(v4f*)(p0)     = t;
    t.x = o0[4]; t.y = o0[5]; t.z = o0[6]; t.w = o0[7]; *(v4f*)(p0 + 4) = t;
    t.x = o1[0]; t.y = o1[1]; t.z = o1[2]; t.w = o1[3]; *(v4f*)(p1)     = t;
    t.x = o1[4]; t.y = o1[5]; t.z = o1[6]; t.w = o1[7]; *(v4f*)(p1 + 4) = t;
  }
}

// ---------------------------------------------------------------------------
extern "C" void kernel_launch(void* const* d_in, const int* in_sizes, int n_in,
                              void* d_out, int out_size, void* d_ws, size_t ws_size,
                              hipStream_t stream) {
  const float* x  = (const float*)d_in[0];
  const float* wq = (const float*)d_in[1];
  const float* wk = (const float*)d_in[2];
  const float* wv = (const float*)d_in[3];
  const float* wp = (const float*)d_in[4];

  float* Qb = (float*)d_ws;                          // [B,64,4096]
  float* Kb = Qb + (size_t)kB * kC * kN;             // [B,64,1024]
  float* Vb = Kb + (size_t)kB * kC * kNkv;           // [B,64,1024]
  float* Ab = Vb + (size_t)kB * kC * kNkv;           // [B,64,4096]

  sra_conv1x1<<<dim3(kN / 256, kB), 256, 0, stream>>>(x, wq, Qb);
  sra_conv2x2s2<<<dim3(kNkv / 256, kB), 256, 0, stream>>>(x, wk, Kb);
  sra_conv2x2s2<<<dim3(kNkv / 256, kB), 256, 0, stream>>>(x, wv, Vb);

  const size_t smem =
      (size_t)(kHd * kNkv + kNkv * kVST + 8 * 16 * kPST) * sizeof(float);
  (void)hipFuncSetAttribute((const void*)sra_attn,
                            hipFuncAttributeMaxDynamicSharedMemorySize,
                            (int)smem);
  sra_attn<<<dim3(kN / 128, kHeads, kB), 256, smem, stream>>>(Qb, Kb, Vb, Ab);

  sra_conv1x1<<<dim3(kN / 256, kB), 256, 0, stream>>>(Ab, wp, (float*)d_out);
}